// SVDTextureTransformBlock_80109730005547
// MI455X (gfx1250) — compile-verified
//
#include <hip/hip_runtime.h>
#include <hip/hip_bf16.h>

typedef float v2f __attribute__((ext_vector_type(2)));
typedef float v8f __attribute__((ext_vector_type(8)));

#define WMMA_F32(a, b, c)                                                    \
  __builtin_amdgcn_wmma_f32_16x16x4_f32(false, (a), false, (b), (short)0,    \
                                        (c), false, false)

#define EPS_COV 1e-5f
#define NB 8
#define NC 256
#define HP 4096            // pooled pixels per (n,c)
#define HWF 16384          // full pixels per (n,c)
#define MATSZ 65536        // 256*256

__device__ __forceinline__ unsigned enc_ord(float f) {
  unsigned u = __float_as_uint(f);
  return (u & 0x80000000u) ? ~u : (u | 0x80000000u);
}
__device__ __forceinline__ float dec_ord(unsigned u) {
  return (u & 0x80000000u) ? __uint_as_float(u ^ 0x80000000u)
                           : __uint_as_float(~u);
}

// ---------------- K0: init stats ----------------
__global__ void init_stats_kernel(float* SB, unsigned* MM) {
  int t = threadIdx.x;
  if (t < 48) SB[t] = 0.0f;
  if (t == 0) MM[0] = 0xFFFFFFFFu;  // min sentinel (encoded space)
  if (t == 1) MM[1] = 0x00000000u;  // max sentinel
}

// ---------------- K1: pool + per-(n,c) mean + global min/max ----------------
__global__ void pool_kernel(const float* __restrict__ x, float* __restrict__ P,
                            float* __restrict__ means, unsigned* __restrict__ MM) {
  int nc = blockIdx.x;          // 0..2047
  int t = threadIdx.x;          // 0..255
  const float* xi = x + (size_t)nc * HWF;
  float* Pi = P + (size_t)nc * HP;
  float sum = 0.f, vmin = 3.4e38f, vmax = -3.4e38f;
  for (int e = t; e < HP; e += 256) {
    int py = e >> 6, px = e & 63;
    const float* r0 = xi + (size_t)(py * 2) * 128 + px * 2;
    v2f top = *(const v2f*)r0;
    v2f bot = *(const v2f*)(r0 + 128);
    float p = (top.x + top.y + bot.x + bot.y) * 0.25f;
    Pi[e] = p;
    sum += p;
    vmin = fminf(vmin, p);
    vmax = fmaxf(vmax, p);
  }
  __shared__ float ss[256], sn[256], sx[256];
  ss[t] = sum; sn[t] = vmin; sx[t] = vmax;
  __syncthreads();
  for (int w = 128; w > 0; w >>= 1) {
    if (t < w) {
      ss[t] += ss[t + w];
      sn[t] = fminf(sn[t], sn[t + w]);
      sx[t] = fmaxf(sx[t], sx[t + w]);
    }
    __syncthreads();
  }
  if (t == 0) {
    means[nc] = ss[0] * (1.0f / 4096.0f);
    atomicMin(&MM[0], enc_ord(sn[0]));
    atomicMax(&MM[1], enc_ord(sx[0]));
  }
}

// ---------------- K2: cov = s*(P P^T - 4096 m m^T) + eps I ----------------
__global__ void __launch_bounds__(32)
cov_kernel(const float* __restrict__ P, const float* __restrict__ means,
           float* __restrict__ cov) {
  int n = blockIdx.y;
  int ti = blockIdx.x >> 4, tj = blockIdx.x & 15;
  int l = threadIdx.x, lr = l & 15, h = l >> 4;
  const float* Pn = P + (size_t)n * NC * HP;
  const float* Arow = Pn + (size_t)(ti * 16 + lr) * HP + 2 * h;
  const float* Brow = Pn + (size_t)(tj * 16 + lr) * HP + 2 * h;
  v8f acc = {};
  for (int k = 0; k < HP; k += 4) {
    v2f a = *(const v2f*)(Arow + k);
    v2f b = *(const v2f*)(Brow + k);
    acc = WMMA_F32(a, b, acc);
  }
  const float s = 4.0f / 16383.0f;
  const float s4096 = s * 4096.0f;
  const float* mb = means + n * NC;
  int col = tj * 16 + lr;
  float mc = mb[col];
  float* Cn = cov + (size_t)n * MATSZ;
#pragma unroll
  for (int v = 0; v < 8; ++v) {
    int row = ti * 16 + v + 8 * h;
    float val = s * acc[v] - s4096 * mb[row] * mc + (row == col ? EPS_COV : 0.f);
    Cn[(size_t)row * NC + col] = val;
  }
}

// ---------------- K3: tau = trace(cov), store 1/tau and rsqrt(tau) ----------
__global__ void trace_kernel(const float* __restrict__ cov, float* SB) {
  int n = blockIdx.x, t = threadIdx.x;
  __shared__ float s[256];
  s[t] = cov[((size_t)n * NC + t) * NC + t];
  __syncthreads();
  for (int w = 128; w > 0; w >>= 1) {
    if (t < w) s[t] += s[t + w];
    __syncthreads();
  }
  if (t == 0) {
    float tau = s[0];
    SB[n] = 1.0f / tau;       // invTau
    SB[8 + n] = rsqrtf(tau);  // tau^{-1/2}
  }
}

// ---------------- K4: Y0 = cov/tau, Z0 = I ----------------
__global__ void init_yz_kernel(const float* __restrict__ cov,
                               const float* __restrict__ SB,
                               float* __restrict__ Y, float* __restrict__ Z) {
  size_t idx = (size_t)blockIdx.x * 256 + threadIdx.x;  // < 8*65536
  int n = (int)(idx >> 16);
  int ij = (int)(idx & 65535);
  Y[idx] = cov[idx] * SB[n];
  Z[idx] = ((ij >> 8) == (ij & 255)) ? 1.0f : 0.0f;
}

// ---- generic batched 256^3 GEMM, B SYMMETRIC: C = alpha*A@B + betaI*I ----
// All B operands here (NS / Newton iterates) are polynomials of one symmetric
// matrix, hence symmetric: B[k][col] == B[col][k], so the lane's B fragment
// is a contiguous, 8B-aligned float2 at B[col][k0..k0+1] -> one b64 load.
__global__ void __launch_bounds__(32)
gemm256_kernel(const float* __restrict__ A, const float* __restrict__ B,
               float* __restrict__ C, float alpha, float betaI) {
  int n = blockIdx.y;
  int ti = blockIdx.x >> 4, tj = blockIdx.x & 15;
  int l = threadIdx.x, lr = l & 15, h = l >> 4;
  const float* An = A + (size_t)n * MATSZ;
  const float* Bn = B + (size_t)n * MATSZ;
  const float* Arow = An + (size_t)(ti * 16 + lr) * NC + 2 * h;
  const float* Brow = Bn + (size_t)(tj * 16 + lr) * NC + 2 * h;  // symmetric B
  v8f acc = {};
  for (int k = 0; k < NC; k += 4) {
    v2f a = *(const v2f*)(Arow + k);
    v2f b = *(const v2f*)(Brow + k);
    acc = WMMA_F32(a, b, acc);
  }
  float* Cn = C + (size_t)n * MATSZ;
  int col = tj * 16 + lr;
#pragma unroll
  for (int v = 0; v < 8; ++v) {
    int row = ti * 16 + v + 8 * h;
    Cn[(size_t)row * NC + col] = alpha * acc[v] + (row == col ? betaI : 0.f);
  }
}

// ---------------- elementwise scale by per-batch SB[sboff+n] ----------------
__global__ void scale_kernel(const float* __restrict__ In,
                             const float* __restrict__ SB, int sboff,
                             float* __restrict__ Out) {
  size_t idx = (size_t)blockIdx.x * 256 + threadIdx.x;
  int n = (int)(idx >> 16);
  Out[idx] = In[idx] * SB[sboff + n];
}

// ---------------- K7: WH = WM @ (P - m) (M=256,N=4096,K=256) ----------------
__global__ void __launch_bounds__(32)
whiten_kernel(const float* __restrict__ WM, const float* __restrict__ P,
              const float* __restrict__ means, float* __restrict__ WH) {
  int n = blockIdx.y;
  int ti = blockIdx.x >> 8;   // 0..15
  int tj = blockIdx.x & 255;  // 0..255
  int l = threadIdx.x, lr = l & 15, h = l >> 4;
  const float* An = WM + (size_t)n * MATSZ;
  const float* Pn = P + (size_t)n * NC * HP;
  const float* mb = means + n * NC;
  const float* Arow = An + (size_t)(ti * 16 + lr) * NC + 2 * h;
  int colg = tj * 16 + lr;
  v8f acc = {};
  for (int k = 0; k < NC; k += 4) {
    v2f a = *(const v2f*)(Arow + k);
    int k0 = k + 2 * h;
    v2f b;
    b.x = Pn[(size_t)k0 * HP + colg] - mb[k0];
    b.y = Pn[(size_t)(k0 + 1) * HP + colg] - mb[k0 + 1];
    acc = WMMA_F32(a, b, acc);
  }
  float* WHn = WH + (size_t)n * NC * HP;
#pragma unroll
  for (int v = 0; v < 8; ++v) {
    int row = ti * 16 + v + 8 * h;
    WHn[(size_t)row * HP + colg] = acc[v];
  }
}

// ---------------- K8: lower-triangle stats of WM ----------------
__global__ void wmstats_kernel(const float* __restrict__ WM, float* SB) {
  int n = blockIdx.x, t = threadIdx.x;
  const float* W = WM + (size_t)n * MATSZ;
  float s1 = 0.f, s2 = 0.f;
  for (int idx = t; idx < MATSZ; idx += 256) {
    int i = idx >> 8, j = idx & 255;
    if (i > j) {
      float w = W[idx];
      s1 += w;
      s2 += w * w;
    }
  }
  __shared__ float a[256], b[256];
  a[t] = s1; b[t] = s2;
  __syncthreads();
  for (int w = 128; w > 0; w >>= 1) {
    if (t < w) { a[t] += a[t + w]; b[t] += b[t + w]; }
    __syncthreads();
  }
  if (t == 0) {
    const float num = 32640.0f;  // C*(C-1)/2
    float mean = a[0] / num;
    // sum over all C^2 of (wm_trid - mean)^2, divided by num (faithful to ref)
    float var = (b[0] - 2.0f * mean * a[0] + 65536.0f * mean * mean) / num;
    SB[16 + n] = mean;
    SB[24 + n] = 10.0f * sqrtf(var);
  }
}

// ---------------- K9: build wm_s (symmetric, original diagonal) -------------
__global__ void build_wms_kernel(const float* __restrict__ WM,
                                 const float* __restrict__ eps,
                                 const float* __restrict__ SB,
                                 float* __restrict__ WMS) {
  size_t idx = (size_t)blockIdx.x * 256 + threadIdx.x;
  int n = (int)(idx >> 16);
  int ij = (int)(idx & 65535);
  int i = ij >> 8, j = ij & 255;
  float mean = SB[16 + n], s10 = SB[24 + n];
  float v;
  if (i == j)
    v = WM[(size_t)n * MATSZ + (size_t)i * 257];
  else if (i < j)
    v = mean + s10 * eps[idx];
  else
    v = mean + s10 * eps[(size_t)n * MATSZ + (size_t)j * NC + i];
  WMS[idx] = v;
}

// ---------------- K10: Newton scaling: 1/(||A||_inf^2) (symmetric A) --------
__global__ void newton_norm_kernel(const float* __restrict__ WMS, float* SB) {
  int n = blockIdx.x, t = threadIdx.x;
  const float* W = WMS + (size_t)n * MATSZ + (size_t)t * NC;
  float r = 0.f;
  for (int j = 0; j < NC; ++j) r += fabsf(W[j]);
  __shared__ float s[256];
  s[t] = r;
  __syncthreads();
  for (int w = 128; w > 0; w >>= 1) {
    if (t < w) s[t] = fmaxf(s[t], s[t + w]);
    __syncthreads();
  }
  if (t == 0) SB[32 + n] = 1.0f / (s[0] * s[0]);
}

// ---------------- K12: target = XN@WH, + mean, clip, 2x2 expand, + HH -------
__global__ void __launch_bounds__(32)
target_fuse_kernel(const float* __restrict__ XN, const float* __restrict__ WH,
                   const float* __restrict__ P, const float* __restrict__ x,
                   const float* __restrict__ means,
                   const unsigned* __restrict__ MM, float* __restrict__ out) {
  int n = blockIdx.y;
  int ti = blockIdx.x >> 8, tj = blockIdx.x & 255;
  int l = threadIdx.x, lr = l & 15, h = l >> 4;
  float cmin = dec_ord(MM[0]);
  float cmax = dec_ord(MM[1]);
  const float* An = XN + (size_t)n * MATSZ;
  const float* Bn = WH + (size_t)n * NC * HP;
  const float* Arow = An + (size_t)(ti * 16 + lr) * NC + 2 * h;
  int colg = tj * 16 + lr;
  v8f acc = {};
  for (int k = 0; k < NC; k += 4) {
    v2f a = *(const v2f*)(Arow + k);
    int k0 = k + 2 * h;
    v2f b;
    b.x = Bn[(size_t)k0 * HP + colg];
    b.y = Bn[(size_t)(k0 + 1) * HP + colg];
    acc = WMMA_F32(a, b, acc);
  }
  int py = colg >> 6, px = colg & 63;
#pragma unroll
  for (int v = 0; v < 8; ++v) {
    int c = ti * 16 + v + 8 * h;
    size_t ncb = (size_t)n * NC + c;
    float t = acc[v] + means[ncb];
    t = fminf(fmaxf(t, cmin), cmax);
    float pl = P[ncb * HP + colg];
    float add = t - pl;  // target - up(pooled); out = x + add
    size_t xoff = ncb * HWF + (size_t)(py * 2) * 128 + px * 2;
    v2f top = *(const v2f*)(x + xoff);
    v2f bot = *(const v2f*)(x + xoff + 128);
    v2f otop = { top.x + add, top.y + add };
    v2f obot = { bot.x + add, bot.y + add };
    *(v2f*)(out + xoff) = otop;
    *(v2f*)(out + xoff + 128) = obot;
  }
}

extern "C" void kernel_launch(void* const* d_in, const int* in_sizes, int n_in,
                              void* d_out, int out_size, void* d_ws,
                              size_t ws_size, hipStream_t stream) {
  (void)in_sizes; (void)n_in; (void)out_size; (void)ws_size;
  const float* x = (const float*)d_in[0];     // (8,256,128,128)
  const float* eps = (const float*)d_in[1];   // (8,256,256)
  float* out = (float*)d_out;

  const size_t BIG = (size_t)NB * NC * HP;    // 8,388,608 floats
  const size_t MAT = (size_t)NB * MATSZ;      // 524,288 floats
  float* ws = (float*)d_ws;
  float* P   = ws;            // pooled (N,C,4096)
  float* WH  = P + BIG;       // whitened at pooled res
  float* COV = WH + BIG;
  float* Y   = COV + MAT;
  float* Y2  = Y + MAT;
  float* Z   = Y2 + MAT;
  float* Z2  = Z + MAT;
  float* T   = Z2 + MAT;
  float* WM  = T + MAT;
  float* WMS = WM + MAT;
  float* X   = WMS + MAT;
  float* X2  = X + MAT;
  float* MEANS = X2 + MAT;                    // 2048 floats
  float* SB  = MEANS + 2048;                  // 48 per-batch scalars
  unsigned* MM = (unsigned*)(SB + 48);        // global min/max (encoded)

  hipLaunchKernelGGL(init_stats_kernel, dim3(1), dim3(64), 0, stream, SB, MM);
  hipLaunchKernelGGL(pool_kernel, dim3(NB * NC), dim3(256), 0, stream,
                     x, P, MEANS, MM);
  hipLaunchKernelGGL(cov_kernel, dim3(256, NB), dim3(32), 0, stream,
                     P, MEANS, COV);
  hipLaunchKernelGGL(trace_kernel, dim3(NB), dim3(256), 0, stream, COV, SB);
  hipLaunchKernelGGL(init_yz_kernel, dim3(2048), dim3(256), 0, stream,
                     COV, SB, Y, Z);

  // Newton-Schulz coupled iteration: Z -> (cov/tau)^{-1/2}
  float* Yc = Y; float* Yn = Y2; float* Zc = Z; float* Zn = Z2;
  const int NS_ITERS = 18;
  for (int it = 0; it < NS_ITERS; ++it) {
    // T = 0.5*(3I - Z@Y)
    hipLaunchKernelGGL(gemm256_kernel, dim3(256, NB), dim3(32), 0, stream,
                       Zc, Yc, T, -0.5f, 1.5f);
    hipLaunchKernelGGL(gemm256_kernel, dim3(256, NB), dim3(32), 0, stream,
                       Yc, T, Yn, 1.0f, 0.0f);
    hipLaunchKernelGGL(gemm256_kernel, dim3(256, NB), dim3(32), 0, stream,
                       T, Zc, Zn, 1.0f, 0.0f);
    float* tmp;
    tmp = Yc; Yc = Yn; Yn = tmp;
    tmp = Zc; Zc = Zn; Zn = tmp;
  }
  // WM = Z * tau^{-1/2}
  hipLaunchKernelGGL(scale_kernel, dim3(2048), dim3(256), 0, stream,
                     Zc, SB, 8, WM);
  // WH = WM @ (P - mean)
  hipLaunchKernelGGL(whiten_kernel, dim3(4096, NB), dim3(32), 0, stream,
                     WM, P, MEANS, WH);
  // randomize texture
  hipLaunchKernelGGL(wmstats_kernel, dim3(NB), dim3(256), 0, stream, WM, SB);
  hipLaunchKernelGGL(build_wms_kernel, dim3(2048), dim3(256), 0, stream,
                     WM, eps, SB, WMS);
  // Newton inverse: X -> WMS^{-1}
  hipLaunchKernelGGL(newton_norm_kernel, dim3(NB), dim3(256), 0, stream,
                     WMS, SB);
  hipLaunchKernelGGL(scale_kernel, dim3(2048), dim3(256), 0, stream,
                     WMS, SB, 32, X);  // X0 = A^T/(||A||_1 ||A||_inf), A symmetric
  float* Xc = X; float* Xn = X2;
  const int NEWTON_ITERS = 32;
  for (int it = 0; it < NEWTON_ITERS; ++it) {
    // T = 2I - A@X ; X = X@T
    hipLaunchKernelGGL(gemm256_kernel, dim3(256, NB), dim3(32), 0, stream,
                       WMS, Xc, T, -1.0f, 2.0f);
    hipLaunchKernelGGL(gemm256_kernel, dim3(256, NB), dim3(32), 0, stream,
                       Xc, T, Xn, 1.0f, 0.0f);
    float* tmp = Xc; Xc = Xn; Xn = tmp;
  }
  // out = HH + clip(XN@WH + mean) expanded 2x2
  hipLaunchKernelGGL(target_fuse_kernel, dim3(4096, NB), dim3(32), 0, stream,
                     Xc, WH, P, x, MEANS, MM, out);
}